// CZRing_20624432956334
// MI455X (gfx1250) — compile-verified
//
#include <hip/hip_runtime.h>

// CZ-ring on 12 qubits is a DIAGONAL unitary with entries ±1:
//   d[row] = (-1)^popcount( row & ror12(row, 1) )
// so out[row, k] = d[row] * x[row, k].  Pure streaming kernel:
// 16 MiB in + 16 MiB out -> ~1.4 us at 23.3 TB/s HBM (and the whole
// problem fits in the 192 MB L2, so replays run at L2 bandwidth).
// float4 (b128) loads/stores, wave32, 8 waves/block.

__global__ __launch_bounds__(256) void czring_diag_sign(const float4* __restrict__ x4,
                                                        float4* __restrict__ o4,
                                                        unsigned n4) {
    const unsigned idx4 = blockIdx.x * 256u + threadIdx.x;
    if (idx4 >= n4) return;

    // 1024 floats per row => 256 float4 per row
    const unsigned row = idx4 >> 8;                       // [0, 4096)

    // ring-CZ parity: adjacent-bit pairs of the 12-bit row index (with wrap)
    const unsigned r   = ((row >> 1) | (row << 11)) & 0xFFFu;  // ror within 12 bits
    const unsigned par = __popc(row & r) & 1u;                 // v_bcnt + and
    const unsigned smask = par << 31;                          // sign-bit xor mask

    // prefetch the load stream a few KB ahead (gfx1250 global_prefetch_b8)
    __builtin_prefetch(x4 + idx4 + 512, 0, 1);

    float4 v = x4[idx4];                                  // global_load_b128

    // exact IEEE sign flip == multiply by -1.0f
    unsigned a = __float_as_uint(v.x) ^ smask;
    unsigned b = __float_as_uint(v.y) ^ smask;
    unsigned c = __float_as_uint(v.z) ^ smask;
    unsigned d = __float_as_uint(v.w) ^ smask;
    v.x = __uint_as_float(a);
    v.y = __uint_as_float(b);
    v.z = __uint_as_float(c);
    v.w = __uint_as_float(d);

    o4[idx4] = v;                                         // global_store_b128
}

extern "C" void kernel_launch(void* const* d_in, const int* in_sizes, int n_in,
                              void* d_out, int out_size, void* d_ws, size_t ws_size,
                              hipStream_t stream) {
    (void)n_in; (void)out_size; (void)d_ws; (void)ws_size;

    const float* x = (const float*)d_in[0];   // [4096, 1024] row-major f32
    float* out     = (float*)d_out;           // [4096, 1024] f32

    const unsigned n  = (unsigned)in_sizes[0];   // 4096*1024 = 4,194,304
    const unsigned n4 = n >> 2;                  // float4 count = 1,048,576

    const unsigned threads = 256;
    const unsigned blocks  = (n4 + threads - 1) / threads;  // 4096

    czring_diag_sign<<<blocks, threads, 0, stream>>>(
        reinterpret_cast<const float4*>(x),
        reinterpret_cast<float4*>(out),
        n4);
}